// SimplexWeightScale_81243601371278
// MI455X (gfx1250) — compile-verified
//
#include <hip/hip_runtime.h>
#include <cstdint>

// ---------------------------------------------------------------------------
// SimplexWeightScale (stick-breaking) for MI455X / gfx1250.
// Memory-bound stream (~520 MB @ 23.3 TB/s -> ~22us floor). No matmul
// structure -> WMMA inapplicable; optimize the data-movement path instead:
// TDM tensor_load_to_lds stages each wave's 1920B input chunk into LDS
// (async, TENSORcnt), conflict-free LDS reads (stride 15 dwords, gcd(15,64)=1),
// and 16B-aligned non-temporal b128 stores for the 64B output rows.
// This toolchain exposes the 6-arg builtin:
//   (uint32x4 g0, int32x8 g1, int32x4 g2, int32x4 g3, int32x8 g4, i32 cpol)
// ---------------------------------------------------------------------------

typedef __attribute__((ext_vector_type(4))) unsigned int v4u;
typedef __attribute__((ext_vector_type(8))) int          v8i;
typedef __attribute__((ext_vector_type(4))) int          v4i;
typedef __attribute__((ext_vector_type(4))) float        v4f;

#define K_IN            15
#define K_OUT           16
#define ROWS_PER_BLOCK  256
#define ROWS_PER_WAVE   32

#if defined(__has_builtin)
#if __has_builtin(__builtin_amdgcn_tensor_load_to_lds)
#define HAVE_TDM 1
#endif
#endif

__global__ __launch_bounds__(ROWS_PER_BLOCK)
void SimplexWeightScale_gfx1250_kernel(const float* __restrict__ Y,
                                       float* __restrict__ out,
                                       int n_rows) {
    __shared__ float tile[ROWS_PER_BLOCK * K_IN];   // 15360 B of LDS

    const int lane = threadIdx.x & 31;
    // force wave id into an SGPR so the TDM descriptor is fully scalar
    const int wv = __builtin_amdgcn_readfirstlane((int)(threadIdx.x >> 5));

    const long long blockFirst = (long long)blockIdx.x * ROWS_PER_BLOCK;
    const long long waveFirst  = blockFirst + (long long)wv * ROWS_PER_WAVE;

    int remRows = n_rows - (int)waveFirst;
    if (remRows > ROWS_PER_WAVE) remRows = ROWS_PER_WAVE;

#ifdef HAVE_TDM
    if (remRows > 0) {
        const unsigned elems   = (unsigned)remRows * K_IN;          // dwords
        const unsigned ldsByte = (unsigned)(uintptr_t)(void*)tile
                               + (unsigned)wv * (ROWS_PER_WAVE * K_IN * 4);
        const uint64_t gbyte   = (uint64_t)(uintptr_t)Y
                               + (uint64_t)waveFirst * (K_IN * 4);

        // ---- D# group 0 (128b): count=1 | lds_addr | global_addr | type=2
        v4u g0;
        g0.x = 1u;                                            // count=1, user mode
        g0.y = ldsByte;                                       // lds_addr[31:0]
        g0.z = (unsigned)(gbyte & 0xFFFFFFFFu);               // global_addr[31:0]
        g0.w = (unsigned)((gbyte >> 32) & 0x1FFFFFFu)         // global_addr[56:32]
             | (2u << 30);                                    // type = 2 ("image")

        // ---- D# group 1 (256b): 1-D tile of `elems` dwords
        v8i g1;
        g1[0] = 0x00020000;                  // workgroup_mask=0, data_size=2 (4B)
        g1[1] = (int)((elems & 0xFFFFu) << 16);    // tensor_dim0[15:0]  @bits63:48
        g1[2] = (int)((elems >> 16) | (1u << 16)); // tensor_dim0[31:16], tensor_dim1=1
        g1[3] = (int)(elems << 16);          // tensor_dim1 hi=0, tile_dim0=elems
        g1[4] = 0;                           // tile_dim1=0 (unused), tile_dim2=0
        g1[5] = (int)elems;                  // tensor_dim0_stride[31:0]
        g1[6] = 0;                           // stride0 hi, stride1 lo
        g1[7] = 0;                           // stride1 hi

        v4i g2 = {0, 0, 0, 0};               // <=2-D tensor: groups 2/3 unused
        v4i g3 = {0, 0, 0, 0};
        v8i g4 = {0, 0, 0, 0, 0, 0, 0, 0};   // extra group (unused, zero)

        __builtin_amdgcn_tensor_load_to_lds(g0, g1, g2, g3, g4, 0 /*cpol*/);
        __builtin_amdgcn_s_wait_tensorcnt(0);
    }
#else
    // Fallback path (toolchain without the TDM builtin): direct global->LDS.
    if (remRows > 0) {
        const float* gsrc = Y + waveFirst * K_IN;
        const int nd = remRows * K_IN;
        for (int i = lane; i < nd; i += 32)
            tile[(unsigned)wv * (ROWS_PER_WAVE * K_IN) + i] = gsrc[i];
    }
#endif

    const long long row = waveFirst + lane;
    if (row >= n_rows) return;

    // Conflict-free LDS reads: dword index = 15*lane + i, gcd(15,64)=1.
    const float* lrow = &tile[((unsigned)wv * ROWS_PER_WAVE + (unsigned)lane) * K_IN];

    float y[K_IN];
#pragma unroll
    for (int i = 0; i < K_IN; ++i) y[i] = lrow[i];

    float o[K_OUT];
    float prod = 1.0f;
#pragma unroll
    for (int i = 0; i < K_IN; ++i) {
        o[i]  = y[i] * prod;
        prod *= (1.0f - y[i]);
    }
    o[K_OUT - 1] = prod;

    // 64B output row, 16B aligned: 4x non-temporal b128 stores.
    float* op = out + row * K_OUT;
#pragma unroll
    for (int v = 0; v < 4; ++v) {
        v4f pack = { o[4 * v + 0], o[4 * v + 1], o[4 * v + 2], o[4 * v + 3] };
        __builtin_nontemporal_store(pack, (v4f*)(op + 4 * v));
    }
}

extern "C" void kernel_launch(void* const* d_in, const int* in_sizes, int n_in,
                              void* d_out, int out_size, void* d_ws, size_t ws_size,
                              hipStream_t stream) {
    (void)n_in; (void)out_size; (void)d_ws; (void)ws_size;
    const float* Y   = (const float*)d_in[0];
    float*       out = (float*)d_out;
    const int n_rows = in_sizes[0] / K_IN;            // 4,194,304
    const int blocks = (n_rows + ROWS_PER_BLOCK - 1) / ROWS_PER_BLOCK;
    SimplexWeightScale_gfx1250_kernel<<<blocks, ROWS_PER_BLOCK, 0, stream>>>(Y, out, n_rows);
}